// STAE_22041772163191
// MI455X (gfx1250) — compile-verified
//
#include <hip/hip_runtime.h>

// ---------------------------------------------------------------------------
// STAE (graph-conv GRU autoencoder) for MI455X / gfx1250, wave32, fp32 WMMA.
// B=8, T=12, N=256, K=9 supports, H=128, H_CODE=16, F_REC=2.
// All matmuls run on V_WMMA_F32_16X16X4_F32 (exact fp32 vs reference).
// gconv reassociated:  sum_k A_k @ (X @ W_k)  ==  Acat(256x2304) @ Zstack.
// Weights and the intermediate Z are stored PAIR-INTERLEAVED
// ( [k/4-group][col][row&3] ) so every B-fragment load is a single b64;
// A-fragment loads are v2f (b64).  Big GEMMs: 32x64 tile/wave (RT=2,CT=4)
// -> 8 v_wmma per 6 b64 loads per K-step.  No predication in any GEMM loop.
// ---------------------------------------------------------------------------

typedef __attribute__((ext_vector_type(2))) float v2f;
typedef __attribute__((ext_vector_type(8))) float v8f;

#define BQ   8
#define TQ   12
#define NQ   256
#define KQ   9
#define BN   2048      // B*N rows of the "activation matrix"
#define KN   2304      // K*N contraction length of the aggregation GEMM

__device__ __forceinline__ v8f wmma4(v2f a, v2f b, v8f c) {
  // D(16x16,f32) = A(16x4,f32) x B(4x16,f32) + C
  return __builtin_amdgcn_wmma_f32_16x16x4_f32(
      false, a, false, b, (short)0, c, false, false);
}

__device__ __forceinline__ float actf(float v, int act) {
  if (act == 1) return 1.0f / (1.0f + __expf(-v));   // sigmoid
  if (act == 2) return tanhf(v);                      // tanh
  if (act == 3) return fmaxf(v, 0.0f);                // relu
  return v;
}

// ---------------------------------------------------------------------------
// Weight packer: zero-pad W[k,0:f,0:O] -> pair-interleaved
//   Wp[ ((k*(fpad/4) + fr/4)*Opad + o)*4 + (fr&3) ]
// so a lane's (row k0, row k0+1) B-pair is contiguous (one b64 load).
// ---------------------------------------------------------------------------
__global__ void pack_w(const float* __restrict__ W, int f, int O,
                       int fpad, int Opad, float* __restrict__ Wp) {
  const int tid = blockIdx.x * blockDim.x + threadIdx.x;
  if (tid >= KQ * fpad * Opad) return;
  const int j  = tid & 3;                       // row within group of 4
  const int q  = tid >> 2;
  const int o  = q % Opad;
  const int g  = (q / Opad) % (fpad >> 2);      // f-group
  const int k  = (q / Opad) / (fpad >> 2);
  const int fr = (g << 2) + j;
  Wp[tid] = (fr < f && o < O) ? W[((size_t)k * f + fr) * O + o] : 0.0f;
}

// ---------------------------------------------------------------------------
// Z kernel: Z4[b,k] = XH[b*256+n, 0:fpad] @ Wp[k]  (interleaved in & out).
// One wave = (16*RT) x (16*CT) tile.  grid = (BN/(16*RT), Opad/(16*CT), K).
// ---------------------------------------------------------------------------
template <int CT, int RT>
__global__ void __launch_bounds__(32)
zk_gemm(const float* __restrict__ XH, int fpad,
        const float* __restrict__ Wp, int Opad, float* __restrict__ Z) {
  const int lane = threadIdx.x & 31;
  const int idx  = lane & 15;          // A-row / B-col within tile
  const int kh   = (lane >> 4) << 1;   // K offset for fragments
  const int mh   = (lane >> 4) << 3;   // row offset for D fragment
  const int row0 = blockIdx.x * (16 * RT);
  const int col0 = blockIdx.y * (16 * CT);
  const int k    = blockIdx.z;

  const float* xrow[RT];
#pragma unroll
  for (int s = 0; s < RT; ++s)
    xrow[s] = XH + (size_t)(row0 + s * 16 + idx) * fpad + kh;
  const float* wbase = Wp + (size_t)k * fpad * Opad + (size_t)(col0 + idx) * 4 + kh;

  v8f c[RT][CT];
#pragma unroll
  for (int s = 0; s < RT; ++s)
#pragma unroll
    for (int t = 0; t < CT; ++t) c[s][t] = (v8f){};

  for (int kk = 0; kk < fpad; kk += 4) {
    v2f a[RT];
#pragma unroll
    for (int s = 0; s < RT; ++s) a[s] = *(const v2f*)(xrow[s] + kk);
    const float* w0 = wbase + (size_t)(kk >> 2) * (Opad * 4);
#pragma unroll
    for (int t = 0; t < CT; ++t) {
      const v2f b = *(const v2f*)(w0 + t * 64);
#pragma unroll
      for (int s = 0; s < RT; ++s) c[s][t] = wmma4(a[s], b, c[s][t]);
    }
  }
#pragma unroll
  for (int s = 0; s < RT; ++s)
#pragma unroll
    for (int t = 0; t < CT; ++t)
      for (int r = 0; r < 8; ++r) {
        const int g  = row0 + s * 16 + r + mh;   // global row in [0,2048)
        const int b_ = g >> 8, n = g & 255;
        const int col = col0 + t * 16 + idx;
        Z[(size_t)((b_ * KQ + k) * NQ) * Opad +
          ((size_t)(n >> 2) * Opad + col) * 4 + (n & 3)] = c[s][t][r];
      }
}

// ---------------------------------------------------------------------------
// Aggregation: out[b,n,:] = act([A_0|..|A_8](256x2304) @ Z_b(2304xOpad) + bias)
// Z read in interleaved layout (b64 B loads).  grid = (NQ/(16*RT), ..., B).
// ---------------------------------------------------------------------------
template <int CT, int RT>
__global__ void __launch_bounds__(32)
agg_gemm(const float* __restrict__ G /*graph [K,N,N]*/,
         const float* __restrict__ Z, int Opad, int O,
         const float* __restrict__ bias, float* __restrict__ out, int act) {
  const int lane = threadIdx.x & 31;
  const int idx  = lane & 15;
  const int kh   = (lane >> 4) << 1;
  const int mh   = (lane >> 4) << 3;
  const int row0  = blockIdx.x * (16 * RT);
  const int col0  = blockIdx.y * (16 * CT);
  const int batch = blockIdx.z;

  const float* Zb = Z + (size_t)batch * KN * Opad + (size_t)(col0 + idx) * 4 + kh;

  v8f c[RT][CT];
#pragma unroll
  for (int s = 0; s < RT; ++s)
#pragma unroll
    for (int t = 0; t < CT; ++t) c[s][t] = (v8f){};

  for (int k = 0; k < KQ; ++k) {
    const float* Gk[RT];
#pragma unroll
    for (int s = 0; s < RT; ++s)
      Gk[s] = G + ((size_t)k << 16) + ((row0 + s * 16 + idx) << 8) + kh;
    const float* Zk = Zb + (size_t)k * NQ * Opad;
    for (int m = 0; m < NQ; m += 4) {
      v2f a[RT];
#pragma unroll
      for (int s = 0; s < RT; ++s) a[s] = *(const v2f*)(Gk[s] + m);
      const float* z0 = Zk + (size_t)(m >> 2) * (Opad * 4);
#pragma unroll
      for (int t = 0; t < CT; ++t) {
        const v2f b = *(const v2f*)(z0 + t * 64);
#pragma unroll
        for (int s = 0; s < RT; ++s) c[s][t] = wmma4(a[s], b, c[s][t]);
      }
    }
  }
#pragma unroll
  for (int s = 0; s < RT; ++s)
#pragma unroll
    for (int t = 0; t < CT; ++t)
      for (int r = 0; r < 8; ++r) {
        const int row = row0 + s * 16 + r + mh;
        const int col = col0 + t * 16 + idx;
        const float v = (col < O) ? actf(c[s][t][r] + bias[col], act) : 0.0f;
        out[(size_t)(batch * NQ + row) * Opad + col] = v;
      }
}

// ---------------------------------------------------------------------------
// MLP GEMM: out[0:2048,0:O] = act(X[0:2048,0:f] @ W[f,O] + bias).
// f%4==0; grid = (2048/16, O/(16*CT)).  (Tiny share of total work.)
// ---------------------------------------------------------------------------
template <int CT>
__global__ void __launch_bounds__(32)
mlp_gemm(const float* __restrict__ X, int f,
         const float* __restrict__ W, int O,
         const float* __restrict__ bias, float* __restrict__ out, int act) {
  const int lane = threadIdx.x & 31;
  const int idx  = lane & 15;
  const int kh   = (lane >> 4) << 1;
  const int mh   = (lane >> 4) << 3;
  const int row0 = blockIdx.x << 4;
  const int col0 = blockIdx.y * (16 * CT);

  const float* xrow = X + (size_t)(row0 + idx) * f + kh;
  const float* wcol = W + col0 + idx;

  v8f c[CT];
#pragma unroll
  for (int t = 0; t < CT; ++t) c[t] = (v8f){};

  for (int kk = 0; kk < f; kk += 4) {
    const v2f a = *(const v2f*)(xrow + kk);
    const float* w0 = wcol + (size_t)(kk + kh) * O;
#pragma unroll
    for (int t = 0; t < CT; ++t) {
      v2f b;
      b.x = w0[t * 16];
      b.y = w0[O + t * 16];
      c[t] = wmma4(a, b, c[t]);
    }
  }
#pragma unroll
  for (int t = 0; t < CT; ++t)
    for (int r = 0; r < 8; ++r) {
      const int row = row0 + r + mh;
      const int col = col0 + t * 16 + idx;
      out[(size_t)row * O + col] = actf(c[t][r] + bias[col], act);
    }
}

// ---------------------------------------------------------------------------
// Elementwise helpers.
// ---------------------------------------------------------------------------
// xh[bn, 0:fpad] = [ s1(b,n,0:f1) | s2(bn,0:f2)*(mul?mul:1) | zeros ]
__global__ void concat_xh(float* __restrict__ dst, int f1, int f2, int fpad,
                          const float* __restrict__ s1, int s1b, int s1n, int s1off,
                          const float* __restrict__ s2, int ld2,
                          const float* __restrict__ mul, int ldm) {
  const int tid = blockIdx.x * blockDim.x + threadIdx.x;
  if (tid >= BN * fpad) return;
  const int bn = tid / fpad, cidx = tid % fpad;
  const int b_ = bn >> 8, n = bn & 255;
  float v = 0.0f;
  if (cidx < f1) {
    v = s1[(size_t)s1off + (size_t)b_ * s1b + (size_t)n * s1n + cidx];
  } else if (cidx < f1 + f2) {
    const int j = cidx - f1;
    v = s2[(size_t)bn * ld2 + j];
    if (mul) v *= mul[(size_t)bn * ldm + j];
  }
  dst[(size_t)bn * fpad + cidx] = v;
}

// h = z*h + (1-z)*c ; optionally scatter new h into recon[:,t,:,:2]
__global__ void gru_update(const float* __restrict__ z, const float* __restrict__ c,
                           float* __restrict__ h, int ld, int w,
                           float* __restrict__ out2, int t) {
  const int tid = blockIdx.x * blockDim.x + threadIdx.x;
  if (tid >= BN * w) return;
  const int bn = tid / w, j = tid % w;
  const size_t o = (size_t)bn * ld + j;
  const float zv = z[o], cv = c[o], hv = h[o];
  const float nh = zv * hv + (1.0f - zv) * cv;
  h[o] = nh;
  if (out2) {
    const int b_ = bn >> 8, n = bn & 255;
    out2[((size_t)(b_ * TQ + t) * NQ + n) * 2 + j] = nh;
  }
}

__global__ void relu_k(const float* __restrict__ in, float* __restrict__ out, int n) {
  const int tid = blockIdx.x * blockDim.x + threadIdx.x;
  if (tid < n) out[tid] = fmaxf(in[tid], 0.0f);
}

// ---------------------------------------------------------------------------
// Host orchestration.  Input flattening: dict insertion order at top level;
// dicts flattened by sorted keys (jax tree order): Wc,Wr,Wz,bc,br,bz.
// ---------------------------------------------------------------------------
extern "C" void kernel_launch(void* const* d_in, const int* in_sizes, int n_in,
                              void* d_out, int out_size, void* d_ws, size_t ws_size,
                              hipStream_t stream) {
  const float* ts    = (const float*)d_in[0];   // [8,12,256,2]
  const float* graph = (const float*)d_in[1];   // [9,256,256]
  const float *e0Wc = (const float*)d_in[2],  *e0Wr = (const float*)d_in[3],
              *e0Wz = (const float*)d_in[4],  *e0bc = (const float*)d_in[5],
              *e0br = (const float*)d_in[6],  *e0bz = (const float*)d_in[7];
  const float *e1Wc = (const float*)d_in[8],  *e1Wr = (const float*)d_in[9],
              *e1Wz = (const float*)d_in[10], *e1bc = (const float*)d_in[11],
              *e1br = (const float*)d_in[12], *e1bz = (const float*)d_in[13];
  const float *d0Wc = (const float*)d_in[14], *d0Wr = (const float*)d_in[15],
              *d0Wz = (const float*)d_in[16], *d0bc = (const float*)d_in[17],
              *d0br = (const float*)d_in[18], *d0bz = (const float*)d_in[19];
  const float *d1Wc = (const float*)d_in[20], *d1Wr = (const float*)d_in[21],
              *d1Wz = (const float*)d_in[22], *d1bc = (const float*)d_in[23],
              *d1br = (const float*)d_in[24], *d1bz = (const float*)d_in[25];
  const float *mW1 = (const float*)d_in[26], *mW2 = (const float*)d_in[27],
              *mb1 = (const float*)d_in[28], *mb2 = (const float*)d_in[29];
  const float *nW1 = (const float*)d_in[30], *nW2 = (const float*)d_in[31],
              *nb1 = (const float*)d_in[32], *nb2 = (const float*)d_in[33];
  (void)in_sizes; (void)n_in; (void)out_size; (void)ws_size;

  float* ws   = (float*)d_ws;
  float* xh   = ws;                       // [2048, <=256]
  float* Zb   = xh   + (size_t)BN * 256;  // [8,9,256,128] (interleaved)
  float* gz   = Zb   + (size_t)BQ * KQ * NQ * 128;
  float* gr   = gz   + (size_t)BN * 128;
  float* gc   = gr   + (size_t)BN * 128;
  float* h1   = gc   + (size_t)BN * 128;
  float* h2   = h1   + (size_t)BN * 128;
  float* stb  = h2   + (size_t)BN * 128;  // relu(enc_h), later decoder input st
  float* t64  = stb  + (size_t)BN * 128;  // [2048,64]
  float* code = t64  + (size_t)BN * 64;   // [2048,16]
  float* dh1  = code + (size_t)BN * 16;   // decoder hiddens, padded ld=16
  float* dh2  = dh1  + (size_t)BN * 16;
  // packed (padded + pair-interleaved) weights
  float* pe0z = dh2  + (size_t)BN * 16;          // 9*132*128 each
  float* pe0r = pe0z + (size_t)KQ * 132 * 128;
  float* pe0c = pe0r + (size_t)KQ * 132 * 128;
  float* pe1z = pe0c + (size_t)KQ * 132 * 128;   // 9*256*128 each
  float* pe1r = pe1z + (size_t)KQ * 256 * 128;
  float* pe1c = pe1r + (size_t)KQ * 256 * 128;
  float* pd0z = pe1c + (size_t)KQ * 256 * 128;   // 9*132*16 each
  float* pd0r = pd0z + (size_t)KQ * 132 * 16;
  float* pd0c = pd0r + (size_t)KQ * 132 * 16;
  float* pd1z = pd0c + (size_t)KQ * 132 * 16;    // 9*4*16 each
  float* pd1r = pd1z + (size_t)KQ * 4 * 16;
  float* pd1c = pd1r + (size_t)KQ * 4 * 16;

  hipMemsetAsync(h1,  0, (size_t)BN * 128 * 4, stream);
  hipMemsetAsync(h2,  0, (size_t)BN * 128 * 4, stream);
  hipMemsetAsync(dh1, 0, (size_t)BN * 16 * 4, stream);
  hipMemsetAsync(dh2, 0, (size_t)BN * 16 * 4, stream);

  auto pack = [&](const float* W, int f, int O, int fpad, int Opad, float* Wp) {
    const int total = KQ * fpad * Opad;
    pack_w<<<dim3((total + 255) / 256), dim3(256), 0, stream>>>(W, f, O, fpad, Opad, Wp);
  };
  pack(e0Wz, 130, 128, 132, 128, pe0z);
  pack(e0Wr, 130, 128, 132, 128, pe0r);
  pack(e0Wc, 130, 128, 132, 128, pe0c);
  pack(e1Wz, 256, 128, 256, 128, pe1z);
  pack(e1Wr, 256, 128, 256, 128, pe1r);
  pack(e1Wc, 256, 128, 256, 128, pe1c);
  pack(d0Wz, 130, 2, 132, 16, pd0z);
  pack(d0Wr, 130, 2, 132, 16, pd0r);
  pack(d0Wc, 130, 2, 132, 16, pd0c);
  pack(d1Wz, 4, 2, 4, 16, pd1z);
  pack(d1Wr, 4, 2, 4, 16, pd1r);
  pack(d1Wc, 4, 2, 4, 16, pd1c);

  // gate: Opad==128 -> CT=4,RT=2 (32x64 tile); Opad==16 -> CT=1,RT=1
  auto gate = [&](int fpad, const float* Wp, const float* bias,
                  int O, int Opad, int act, float* out) {
    if (Opad == 128) {
      zk_gemm<4, 2><<<dim3(BN / 32, 2, KQ), dim3(32), 0, stream>>>(xh, fpad, Wp, 128, Zb);
      agg_gemm<4, 2><<<dim3(NQ / 32, 2, BQ), dim3(32), 0, stream>>>(graph, Zb, 128, O, bias, out, act);
    } else {
      zk_gemm<1, 1><<<dim3(BN / 16, 1, KQ), dim3(32), 0, stream>>>(xh, fpad, Wp, 16, Zb);
      agg_gemm<1, 1><<<dim3(NQ / 16, 1, BQ), dim3(32), 0, stream>>>(graph, Zb, 16, O, bias, out, act);
    }
  };
  auto concat = [&](int f1, int f2, int fpad, const float* s1, int s1b, int s1n,
                    int s1off, const float* s2, int ld2, const float* mul, int ldm) {
    const int total = BN * fpad;
    concat_xh<<<dim3((total + 255) / 256), dim3(256), 0, stream>>>(
        xh, f1, f2, fpad, s1, s1b, s1n, s1off, s2, ld2, mul, ldm);
  };
  auto update = [&](float* h, int ld, int w, float* out2, int t) {
    const int total = BN * w;
    gru_update<<<dim3((total + 255) / 256), dim3(256), 0, stream>>>(
        gz, gc, h, ld, w, out2, t);
  };

  // ----------------------------- encoder ---------------------------------
  for (int t = 0; t < TQ; ++t) {
    // layer 0: x = ts[:,t] (f1=2), hidden h1, fpad=132
    concat(2, 128, 132, ts, TQ * NQ * 2, 2, t * NQ * 2, h1, 128, nullptr, 0);
    gate(132, pe0z, e0bz, 128, 128, 1, gz);
    gate(132, pe0r, e0br, 128, 128, 1, gr);
    concat(2, 128, 132, ts, TQ * NQ * 2, 2, t * NQ * 2, h1, 128, gr, 128);
    gate(132, pe0c, e0bc, 128, 128, 2, gc);
    update(h1, 128, 128, nullptr, 0);
    // layer 1: x = h1 (f1=128), hidden h2, fpad=256
    concat(128, 128, 256, h1, NQ * 128, 128, 0, h2, 128, nullptr, 0);
    gate(256, pe1z, e1bz, 128, 128, 1, gz);
    gate(256, pe1r, e1br, 128, 128, 1, gr);
    concat(128, 128, 256, h1, NQ * 128, 128, 0, h2, 128, gr, 128);
    gate(256, pe1c, e1bc, 128, 128, 2, gc);
    update(h2, 128, 128, nullptr, 0);
  }

  // --------------------------- MLP bottleneck ----------------------------
  relu_k<<<dim3((BN * 128 + 255) / 256), dim3(256), 0, stream>>>(h2, stb, BN * 128);
  mlp_gemm<4><<<dim3(BN / 16, 1), dim3(32), 0, stream>>>(stb, 128, mW1, 64, mb1, t64, 3);
  mlp_gemm<1><<<dim3(BN / 16, 1), dim3(32), 0, stream>>>(t64, 64, mW2, 16, mb2, code, 0);
  mlp_gemm<4><<<dim3(BN / 16, 1), dim3(32), 0, stream>>>(code, 16, nW1, 64, nb1, t64, 3);
  mlp_gemm<4><<<dim3(BN / 16, 2), dim3(32), 0, stream>>>(t64, 64, nW2, 128, nb2, stb, 0);

  // ----------------------------- decoder ---------------------------------
  for (int t = 0; t < TQ; ++t) {
    // layer 0: x = st (f1=128), hidden dh1 (w=2, ld=16), fpad=132
    concat(128, 2, 132, stb, NQ * 128, 128, 0, dh1, 16, nullptr, 0);
    gate(132, pd0z, d0bz, 2, 16, 1, gz);
    gate(132, pd0r, d0br, 2, 16, 1, gr);
    concat(128, 2, 132, stb, NQ * 128, 128, 0, dh1, 16, gr, 16);
    gate(132, pd0c, d0bc, 2, 16, 2, gc);
    update(dh1, 16, 2, nullptr, 0);
    // layer 1: x = dh1 (f1=2), hidden dh2, fpad=4; emit recon[:,t]
    concat(2, 2, 4, dh1, NQ * 16, 16, 0, dh2, 16, nullptr, 0);
    gate(4, pd1z, d1bz, 2, 16, 1, gz);
    gate(4, pd1r, d1br, 2, 16, 1, gr);
    concat(2, 2, 4, dh1, NQ * 16, 16, 0, dh2, 16, gr, 16);
    gate(4, pd1c, d1bc, 2, 16, 2, gc);
    update(dh2, 16, 2, (float*)d_out, t);
  }
}